// NormEMAVectorQuantizer_3083786518935
// MI455X (gfx1250) — compile-verified
//
#include <hip/hip_runtime.h>
#include <hip/hip_bf16.h>

// ---------------------------------------------------------------------------
// NormEMA Vector Quantizer for MI455X (gfx1250, wave32)
//   N = 8192 tokens, C = 32, K = 8192 codes
//   cos_sim GEMM+argmax done with V_WMMA_F32_16X16X4_F32 (full f32 precision)
// ---------------------------------------------------------------------------

typedef float v2f __attribute__((ext_vector_type(2)));
typedef float v8f __attribute__((ext_vector_type(8)));

#define N_VEC   8192      // B*H*W
#define K_CODE  8192      // codebook entries
#define CDIM    32
#define DECAYF  0.99f
#define BETAF   1.0f
#define L2EPS   1e-12f

// ---------------- workspace layout (floats) ----------------
// zflat   : [0,              262144)   normalized tokens [N,32]
// tok(int): [262144,         270336)   argmax indices
// bins    : [270336,         278528)
// esum    : [278528,         540672)   scatter-add of zflat per code
// lossAcc : [540672]
// ---------------- d_out layout (floats) --------------------
// zq_out  : [0,      262144)  [B,C,H,W]
// loss    : [262144]
// tok_out : [262145, 270337)  token ids as float [B,H,W]
// new_emb : [270337, 532481)
// new_csz : [532481, 540673)

// -------- kernel 0: zero the scatter-add scratch --------
__global__ void vq_zero(float* __restrict__ bins, float* __restrict__ esum,
                        float* __restrict__ lossAcc) {
    int i = blockIdx.x * blockDim.x + threadIdx.x;
    if (i < K_CODE) bins[i] = 0.0f;
    if (i < K_CODE * CDIM) esum[i] = 0.0f;
    if (i == 0) lossAcc[0] = 0.0f;
}

// -------- kernel 1: per-token L2 normalize (one wave32 per token) --------
// z is [B=32, C=32, H=16, W=16]; token n = b*256 + h*16 + w; lane = channel
__global__ void vq_normalize(const float* __restrict__ z,
                             float* __restrict__ zflat) {
    int wave = (blockIdx.x * blockDim.x + threadIdx.x) >> 5;  // token id
    int lane = threadIdx.x & 31;                              // channel
    int b  = wave >> 8;
    int hw = wave & 255;
    float v = z[b * (CDIM * 256) + lane * 256 + hw];
    float s = v * v;
    #pragma unroll
    for (int off = 16; off > 0; off >>= 1) s += __shfl_xor(s, off, 32);
    float nrm = fmaxf(sqrtf(s), L2EPS);
    zflat[wave * CDIM + lane] = v / nrm;
}

// -------- kernel 2: WMMA cosine-sim + fused argmax --------
// One wave handles 16 tokens.  A = 16x32 tokens, B = 32x16 codes per chunk,
// K=32 done as 8 chained V_WMMA_F32_16X16X4_F32.  Scores never leave VGPRs.
__global__ void vq_argmax(const float* __restrict__ zflat,
                          const float* __restrict__ emb,
                          int* __restrict__ tok,
                          float* __restrict__ tok_out) {
    int wave = (blockIdx.x * blockDim.x + threadIdx.x) >> 5;
    int lane = threadIdx.x & 31;
    int half = lane >> 4;        // K-half selector (A) / row-half (D)
    int m16  = lane & 15;        // A row index / B column index
    int tokenBase = wave * 16;

    // A-matrix (16x4 f32 layout): lane = 16*(k>=2)+m, vgpr = k&1
    //   -> lane loads float2 at  row*32 + 4*kk + 2*half
    v2f a[8];
    const float* arow = zflat + (tokenBase + m16) * CDIM + 2 * half;
    #pragma unroll
    for (int kk = 0; kk < 8; ++kk)
        a[kk] = *(const v2f*)(arow + 4 * kk);

    float best[8];
    int   bidx[8];
    #pragma unroll
    for (int r = 0; r < 8; ++r) { best[r] = -3.402823e38f; bidx[r] = 0; }

    for (int cb = 0; cb < K_CODE; cb += 16) {
        // B-matrix (4x16): mirrored layout, n = lane&15, same float2 pattern,
        // row = codebook row (embedding is row-major [K,32], rows unit-norm)
        const float* brow = emb + (cb + m16) * CDIM + 2 * half;
        v2f bm[8];
        #pragma unroll
        for (int kk = 0; kk < 8; ++kk)
            bm[kk] = *(const v2f*)(brow + 4 * kk);

        // prefetch next chunk's codebook rows (global_prefetch_b8)
        if (cb + 16 < K_CODE)
            __builtin_prefetch(brow + 16 * CDIM, 0, 0);

        v8f acc = {};
        #pragma unroll
        for (int kk = 0; kk < 8; ++kk)
            acc = __builtin_amdgcn_wmma_f32_16x16x4_f32(
                      false, a[kk], false, bm[kk], (short)0, acc, false, false);

        // D layout: lane l, vgpr r -> row (r + 8*half), col (l&15)
        int code = cb + m16;
        #pragma unroll
        for (int r = 0; r < 8; ++r) {
            float s = acc[r];
            bool gt = s > best[r];              // strict > => earliest index wins
            bidx[r] = gt ? code : bidx[r];
            best[r] = gt ? s : best[r];
        }
    }

    // reduce across the 16 lanes of each half (columns); prefer lower index on tie
    #pragma unroll
    for (int off = 1; off < 16; off <<= 1) {
        #pragma unroll
        for (int r = 0; r < 8; ++r) {
            float ob = __shfl_xor(best[r], off, 32);
            int   oi = __shfl_xor(bidx[r], off, 32);
            bool take = (ob > best[r]) || (ob == best[r] && oi < bidx[r]);
            best[r] = take ? ob : best[r];
            bidx[r] = take ? oi : bidx[r];
        }
    }

    if (m16 == 0) {
        #pragma unroll
        for (int r = 0; r < 8; ++r) {
            int row = tokenBase + half * 8 + r;
            tok[row]     = bidx[r];
            tok_out[row] = (float)bidx[r];
        }
    }
}

// -------- kernel 3: gather + straight-through + loss + EMA scatter --------
__global__ void vq_gather(const float* __restrict__ zflat,
                          const float* __restrict__ emb,
                          const int* __restrict__ tok,
                          float* __restrict__ zq_out,
                          float* __restrict__ bins,
                          float* __restrict__ esum,
                          float* __restrict__ lossAcc) {
    int n = blockIdx.x * blockDim.x + threadIdx.x;   // token
    int t = tok[n];
    int b = n >> 8, hw = n & 255;
    float sq = 0.0f;
    #pragma unroll
    for (int c = 0; c < CDIM; ++c) {
        float zn = zflat[n * CDIM + c];
        float e  = emb[t * CDIM + c];
        float d  = e - zn;
        sq += d * d;
        // STE value computed exactly as reference: zn + (z_q - zn)
        zq_out[b * (CDIM * 256) + c * 256 + hw] = zn + d;
        atomicAdd(&esum[t * CDIM + c], zn);
    }
    atomicAdd(&bins[t], 1.0f);

    __shared__ float red[256];
    red[threadIdx.x] = sq;
    __syncthreads();
    for (int s = 128; s > 0; s >>= 1) {
        if (threadIdx.x < s) red[threadIdx.x] += red[threadIdx.x + s];
        __syncthreads();
    }
    if (threadIdx.x == 0) atomicAdd(lossAcc, red[0]);
}

// -------- kernel 4: EMA codebook update + loss finalize --------
__global__ void vq_update(const float* __restrict__ emb,
                          const float* __restrict__ csz,
                          const float* __restrict__ bins,
                          const float* __restrict__ esum,
                          float* __restrict__ new_emb,
                          float* __restrict__ new_csz,
                          float* __restrict__ loss_out,
                          const float* __restrict__ lossAcc) {
    int k = blockIdx.x * blockDim.x + threadIdx.x;   // code id
    if (k == 0)
        loss_out[0] = BETAF * lossAcc[0] / (float)(N_VEC * CDIM);

    float bn   = bins[k];
    new_csz[k] = csz[k] * DECAYF + (1.0f - DECAYF) * bn;
    bool zero  = (bn == 0.0f);
    float bc   = zero ? 1.0f : bn;

    // pass 1: norm of (esum / bc)
    float ss = 0.0f;
    for (int c = 0; c < CDIM; ++c) {
        float x = esum[k * CDIM + c] / bc;
        ss += x * x;
    }
    float nrm = fmaxf(sqrtf(ss), L2EPS);

    // pass 2: norm of ema-blended vector
    float ss2 = 0.0f;
    for (int c = 0; c < CDIM; ++c) {
        float e  = emb[k * CDIM + c];
        float en = zero ? e : (esum[k * CDIM + c] / bc) / nrm;
        float u  = e * DECAYF + (1.0f - DECAYF) * en;
        ss2 += u * u;
    }
    float nrm2 = fmaxf(sqrtf(ss2), L2EPS);

    // pass 3: write
    for (int c = 0; c < CDIM; ++c) {
        float e  = emb[k * CDIM + c];
        float en = zero ? e : (esum[k * CDIM + c] / bc) / nrm;
        float u  = e * DECAYF + (1.0f - DECAYF) * en;
        new_emb[k * CDIM + c] = u / nrm2;
    }
}

extern "C" void kernel_launch(void* const* d_in, const int* in_sizes, int n_in,
                              void* d_out, int out_size, void* d_ws, size_t ws_size,
                              hipStream_t stream) {
    const float* z    = (const float*)d_in[0];   // [32,32,16,16]
    const float* emb  = (const float*)d_in[1];   // [8192,32]
    const float* csz  = (const float*)d_in[2];   // [8192]

    float* ws    = (float*)d_ws;
    float* zflat = ws;                                   // 262144
    int*   tok   = (int*)(ws + 262144);                  // 8192
    float* bins  = ws + 262144 + 8192;                   // 8192
    float* esum  = ws + 262144 + 16384;                  // 262144
    float* lossA = ws + 262144 + 16384 + 262144;         // 1

    float* out      = (float*)d_out;
    float* zq_out   = out;                // 262144
    float* loss_out = out + 262144;       // 1
    float* tok_out  = out + 262145;       // 8192
    float* new_emb  = out + 270337;       // 262144
    float* new_csz  = out + 532481;       // 8192

    // 0: zero scatter scratch (262144 elems dominate)
    vq_zero<<<(K_CODE * CDIM + 255) / 256, 256, 0, stream>>>(bins, esum, lossA);

    // 1: normalize: one wave per token -> 8192 waves -> 1024 blocks of 256
    vq_normalize<<<(N_VEC * 32) / 256, 256, 0, stream>>>(z, zflat);

    // 2: WMMA argmax: one wave per 16 tokens -> 512 waves -> 64 blocks of 256
    vq_argmax<<<(N_VEC / 16) * 32 / 256, 256, 0, stream>>>(zflat, emb, tok, tok_out);

    // 3: gather/STE/loss/scatter: one thread per token
    vq_gather<<<N_VEC / 256, 256, 0, stream>>>(zflat, emb, tok, zq_out,
                                               bins, esum, lossA);

    // 4: EMA update: one thread per code
    vq_update<<<K_CODE / 256, 256, 0, stream>>>(emb, csz, bins, esum,
                                                new_emb, new_csz, loss_out, lossA);
}